// Normalized_Correlation_Layer_64570538328343
// MI455X (gfx1250) — compile-verified
//
#include <hip/hip_runtime.h>

typedef __attribute__((ext_vector_type(2))) float v2f;
typedef __attribute__((ext_vector_type(8))) float v8f;

#define NB   16
#define NR   37
#define NC   12
#define NCH  128
#define NWIN 60
#define CHUNK 32
#define SCOLS 16        // 12 cols + 2-col zero halo each side
#define S1ROWS 5        // unpadded rows r-2..r+2
#define S2ROWS 9        // unpadded rows r-4..r+4
#define S1STRIDE 81     // 5*16, +1 -> odd, conflict-free per-channel reads
#define S2STRIDE 145    // 9*16, +1
#define STSTRIDE 73     // 72 windows, +1

__device__ __forceinline__ int row2_map(int k) {
  // row2 = concat([0,1], arange(37)+2, [37,38]) -> identity for k<=38, then 37,38
  return (k <= 38) ? k : ((k == 39) ? 37 : 38);
}

__global__ __launch_bounds__(256)
void ncorr_wmma_kernel(const float* __restrict__ in1,
                       const float* __restrict__ in2,
                       float* __restrict__ out) {
  __shared__ float lds1[CHUNK * S1STRIDE];
  __shared__ float lds2[CHUNK * S2STRIDE];
  __shared__ float smean[CHUNK * STSTRIDE];
  __shared__ float srstd[CHUNK * STSTRIDE];

  const int bid   = blockIdx.x;
  const int chunk = bid & 3;          // 4 channel chunks of 32
  const int br    = bid >> 2;
  const int b     = br / NR;
  const int r     = br - b * NR;
  const int c0    = chunk * CHUNK;

  const int tid  = threadIdx.x;
  const int wave = tid >> 5;
  const int lane = tid & 31;

  // CDNA5 prefetch of the slab bases (global_prefetch_b8)
  {
    int pr1 = (r >= 2) ? (r - 2) : 0;
    __builtin_prefetch(&in1[((size_t)(b * NR + pr1) * NC) * NCH + c0], 0, 1);
    int pr2 = (r >= 4) ? (r - 4) : 0;
    __builtin_prefetch(&in2[((size_t)(b * NR + pr2) * NC) * NCH + c0], 0, 1);
  }

  // ---- Stage slab1 with materialized zero halo: 5 rows x 16 cols x 32 ch.
  for (int e = tid; e < S1ROWS * SCOLS * CHUNK; e += 256) {
    int cl   = e & 31;
    int rc   = e >> 5;
    int colS = rc & 15;          // storage col 0..15 == unpadded col-2..13
    int i    = rc >> 4;          // row 0..4
    int ucol = colS - 2;
    int grow = r - 2 + i;
    float v = 0.f;
    if (grow >= 0 && grow < NR && ucol >= 0 && ucol < NC)
      v = in1[(((size_t)(b * NR + grow) * NC) + ucol) * NCH + c0 + cl];
    lds1[cl * S1STRIDE + i * SCOLS + colS] = v;
  }
  // ---- Stage slab2: 9 rows x 16 cols x 32 ch.
  for (int e = tid; e < S2ROWS * SCOLS * CHUNK; e += 256) {
    int cl   = e & 31;
    int rc   = e >> 5;
    int colS = rc & 15;
    int i    = rc >> 4;          // row 0..8
    int ucol = colS - 2;
    int grow = r - 4 + i;
    float v = 0.f;
    if (grow >= 0 && grow < NR && ucol >= 0 && ucol < NC)
      v = in2[(((size_t)(b * NR + grow) * NC) + ucol) * NCH + c0 + cl];
    lds2[cl * S2STRIDE + i * SCOLS + colS] = v;
  }
  __syncthreads();

  // ---- Cooperative stats: 72 windows x 32 channels, 9 per thread.
  // Window id is wave-uniform (e>>5), channel = lane -> no divergence,
  // stride-81/145 reads are bank-conflict free. Halo zeros make reads
  // unconditional; zeros participate in mean/std exactly as in jnp.pad.
  for (int e = tid; e < 72 * CHUNK; e += 256) {
    int cl  = e & 31;
    int wid = e >> 5;            // uniform within a wave
    float s = 0.f, s2 = 0.f;
    if (wid < 12) {
      const float* __restrict__ L1 = &lds1[cl * S1STRIDE];
      const int j = wid;
      #pragma unroll
      for (int i = 0; i < 5; ++i) {
        #pragma unroll
        for (int kk = 0; kk < 5; ++kk) {
          float x = L1[i * SCOLS + j + kk];   // storage col = (j-2+kk)+2
          s += x; s2 += x * x;
        }
      }
    } else {
      const float* __restrict__ L2 = &lds2[cl * S2STRIDE];
      const int w = wid - 12;
      const int d = w / 12, j2 = w - d * 12;
      const int rbase = row2_map(r + d) - r;  // 0..4 always
      #pragma unroll
      for (int i = 0; i < 5; ++i) {
        #pragma unroll
        for (int kk = 0; kk < 5; ++kk) {
          float x = L2[(rbase + i) * SCOLS + j2 + kk];
          s += x; s2 += x * x;
        }
      }
    }
    float m   = s * 0.04f;                    // mean over 25 (zeros included)
    float var = s2 * 0.04f - m * m;           // population variance
    smean[cl * STSTRIDE + wid] = m;
    srstd[cl * STSTRIDE + wid] = rsqrtf(var);
  }
  __syncthreads();

  // ---- WMMA phase. Per ISA f32 16x16x4 A/B layout:
  // lanes 0-15 hold K{0,1} in {v0,v1}; lanes 16-31 hold K{2,3}.
  const int m_row = lane & 15;
  const int khalf = lane >> 4;
  const int mrc   = (m_row < 12) ? m_row : 11;     // clamped address row
  const float m6  = (khalf == 0) ? 1.f : 0.f;      // k=6: only p=24 is live

  // Per-lane LDS offsets for the two K slots of each of 7 K-steps
  // (p = 4k + 2*khalf + u, clamped to 24 for the masked tail).
  int aoff0[7], aoff1[7];
  #pragma unroll
  for (int k = 0; k < 7; ++k) {
    int p0  = 4 * k + 2 * khalf;
    int pc0 = (p0 < 25) ? p0 : 24;
    int i0  = pc0 / 5;
    aoff0[k] = i0 * SCOLS + (pc0 - i0 * 5);
    int p1  = (k < 6) ? (p0 + 1) : 24;             // k=6,u=1 is always zero
    int i1  = p1 / 5;
    aoff1[k] = i1 * SCOLS + (p1 - i1 * 5);
  }

  for (int q = 0; q < 4; ++q) {
    const int cl = wave * 4 + q;                   // this wave's channel
    const float* __restrict__ L1 = &lds1[cl * S1STRIDE];
    const float* __restrict__ L2 = &lds2[cl * S2STRIDE];
    const float* __restrict__ SM = &smean[cl * STSTRIDE];
    const float* __restrict__ SR = &srstd[cl * STSTRIDE];

    // A operand (normalized xc1), hoisted over the 4 N tiles.
    const float mA = SM[mrc];
    float rA = SR[mrc];
    rA = (m_row < 12) ? rA : 0.f;   // kills padded rows branchlessly

    v2f a[7];
    #pragma unroll
    for (int k = 0; k < 7; ++k) {
      float x0 = (L1[aoff0[k] + mrc] - mA) * rA;
      if (k == 6) x0 *= m6;
      float x1v = 0.f;
      if (k < 6)
        x1v = (L1[aoff1[k] + mrc] - mA) * rA;
      a[k].x = x0; a[k].y = x1v;
    }

    #pragma unroll
    for (int t = 0; t < 4; ++t) {
      const int w  = 16 * t + m_row;
      const int wc = (w < NWIN) ? w : (NWIN - 1);  // clamped address
      const int d  = wc / 12;
      const int j2 = wc - d * 12;
      const int rbase = row2_map(r + d) - r;
      const int bbase = rbase * SCOLS + j2;
      const float mB = SM[12 + wc];
      float rB = SR[12 + wc];
      rB = (w < NWIN) ? rB : 0.f;                  // kills padded w lanes

      v8f acc = {};
      #pragma unroll
      for (int k = 0; k < 7; ++k) {
        float x0 = (L2[bbase + aoff0[k]] - mB) * rB;
        if (k == 6) x0 *= m6;
        float x1v = 0.f;
        if (k < 6)
          x1v = (L2[bbase + aoff1[k]] - mB) * rB;
        v2f bb; bb.x = x0; bb.y = x1v;
        acc = __builtin_amdgcn_wmma_f32_16x16x4_f32(
            false, a[k], false, bb, (short)0, acc, false, false);
      }

      // D layout: VGPR g -> row g (lanes 0-15) / g+8 (lanes 16-31), col lane&15
      if (w < NWIN) {
        const int c = c0 + cl;
        float* op = out + ((size_t)(b * NR + r) * NC) * (NCH * NWIN)
                        + (size_t)c * NWIN + w;
        #pragma unroll
        for (int g = 0; g < 8; ++g) {
          int j = g + 8 * khalf;
          if (j < 12)
            op[(size_t)j * (NCH * NWIN)] = acc[g];
        }
      }
    }
  }
}

extern "C" void kernel_launch(void* const* d_in, const int* in_sizes, int n_in,
                              void* d_out, int out_size, void* d_ws, size_t ws_size,
                              hipStream_t stream) {
  const float* in1 = (const float*)d_in[0];
  const float* in2 = (const float*)d_in[1];
  float* out = (float*)d_out;
  dim3 grid(NB * NR * 4);   // (b, r, channel-quarter) = 2368 workgroups
  dim3 block(256);          // 8 waves (wave32)
  ncorr_wmma_kernel<<<grid, block, 0, stream>>>(in1, in2, out);
}